// PrototypeLayer_88776974008414
// MI455X (gfx1250) — compile-verified
//
#include <hip/hip_runtime.h>

#define C_CLASSES 1000
#define D_DIM     256
#define EPS       1e-8f

typedef float f32x4 __attribute__((ext_vector_type(4)));
typedef float v2f   __attribute__((ext_vector_type(2)));
typedef float v8f   __attribute__((ext_vector_type(8)));

// ---------------- Kernel 0: zero accumulator (d_out) + conf_sum (d_ws) ----------------
__global__ __launch_bounds__(256) void proto_zero_kernel(float* __restrict__ out,
                                                         float* __restrict__ confsum) {
    const int stride = gridDim.x * blockDim.x;
    int i = blockIdx.x * blockDim.x + threadIdx.x;
    for (int j = i; j < C_CLASSES * D_DIM; j += stride) out[j] = 0.0f;
    for (int j = i; j < C_CLASSES; j += stride) confsum[j] = 0.0f;
}

// ---------------- Kernel 1: streaming scatter-accumulate -----------------------------
// One wave32 per sample: lane l loads float4 at dims [4l..4l+3] and [128+4l..128+4l+3]
// (two fully coalesced b128 loads, non-temporal: the 1GB stream has zero reuse),
// scales by conf, and issues 8 global_atomic_add_f32 into out[label,:] (lives in L2).
__global__ __launch_bounds__(256) void proto_accum_kernel(const float* __restrict__ emb,
                                                          const int*   __restrict__ labels,
                                                          const float* __restrict__ conf,
                                                          float* __restrict__ out,
                                                          float* __restrict__ confsum,
                                                          int n) {
    const int lane  = threadIdx.x & 31;
    const int gwave = (blockIdx.x * blockDim.x + threadIdx.x) >> 5;
    const int nwave = (gridDim.x * blockDim.x) >> 5;
    const f32x4* emb4 = (const f32x4*)emb;

    for (int s = gwave; s < n; s += nwave) {
        // label/conf are wave-uniform -> pull into SGPRs for scalar address math
        const int   lbl = __builtin_amdgcn_readfirstlane(labels[s]);
        const float cw  = conf[s];

        const size_t row = (size_t)s * (D_DIM / 4);
        f32x4 a = __builtin_nontemporal_load(&emb4[row + lane]);        // dims 4l..4l+3
        f32x4 b = __builtin_nontemporal_load(&emb4[row + 32 + lane]);   // dims 128+4l..

        float* dst = out + (size_t)lbl * D_DIM;
        const int o0 = lane * 4;
        const int o1 = 128 + lane * 4;
        __hip_atomic_fetch_add(&dst[o0 + 0], a.x * cw, __ATOMIC_RELAXED, __HIP_MEMORY_SCOPE_AGENT);
        __hip_atomic_fetch_add(&dst[o0 + 1], a.y * cw, __ATOMIC_RELAXED, __HIP_MEMORY_SCOPE_AGENT);
        __hip_atomic_fetch_add(&dst[o0 + 2], a.z * cw, __ATOMIC_RELAXED, __HIP_MEMORY_SCOPE_AGENT);
        __hip_atomic_fetch_add(&dst[o0 + 3], a.w * cw, __ATOMIC_RELAXED, __HIP_MEMORY_SCOPE_AGENT);
        __hip_atomic_fetch_add(&dst[o1 + 0], b.x * cw, __ATOMIC_RELAXED, __HIP_MEMORY_SCOPE_AGENT);
        __hip_atomic_fetch_add(&dst[o1 + 1], b.y * cw, __ATOMIC_RELAXED, __HIP_MEMORY_SCOPE_AGENT);
        __hip_atomic_fetch_add(&dst[o1 + 2], b.z * cw, __ATOMIC_RELAXED, __HIP_MEMORY_SCOPE_AGENT);
        __hip_atomic_fetch_add(&dst[o1 + 3], b.w * cw, __ATOMIC_RELAXED, __HIP_MEMORY_SCOPE_AGENT);
        if (lane == 0)
            __hip_atomic_fetch_add(&confsum[lbl], cw, __ATOMIC_RELAXED, __HIP_MEMORY_SCOPE_AGENT);
    }
}

// ---------------- Kernel 2: normalize via WMMA diagonal scale ------------------------
// out_tile(16x16) = diag(1/(confsum+eps)) (16x16) x proto_tile(16x16), built from
// 4 chained V_WMMA_F32_16X16X4_F32 (K-chunks of the diagonal). One block per 16 classes;
// wave w handles column tiles {w, w+8}. Lane layouts per CDNA5 ISA:
//   A 16x4: vgpr0 = (K=0 | K=2), vgpr1 = (K=1 | K=3); M = lane%16
//   C/D 16x16: vgpr r holds row M=r (lanes 0-15) / M=r+8 (lanes 16-31), N = lane%16
__global__ __launch_bounds__(256) void proto_norm_kernel(float* __restrict__ out,
                                                         const float* __restrict__ confsum) {
    const int lane = threadIdx.x & 31;
    const int wave = threadIdx.x >> 5;
    const int m    = lane & 15;
    const int hi   = (lane >= 16) ? 1 : 0;
    const int base = blockIdx.x * 16;

    int cls = base + m;
    cls = cls > (C_CLASSES - 1) ? (C_CLASSES - 1) : cls;     // select, keeps EXEC full
    const float inv = 1.0f / (confsum[cls] + EPS);

    for (int t = wave; t < 16; t += 8) {
        const int n0 = t * 16;
        v8f acc = {0.f, 0.f, 0.f, 0.f, 0.f, 0.f, 0.f, 0.f};
#pragma unroll
        for (int kk = 0; kk < 4; ++kk) {
            const int ka0 = 4 * kk + (hi ? 2 : 0);           // K col of A vgpr0 for this lane
            const int ka1 = 4 * kk + (hi ? 3 : 1);           // K col of A vgpr1
            v2f A;
            A.x = (m == ka0) ? inv : 0.0f;                   // diag(inv) slice
            A.y = (m == ka1) ? inv : 0.0f;

            int r0 = base + ka0; r0 = r0 > (C_CLASSES - 1) ? (C_CLASSES - 1) : r0;
            int r1 = base + ka1; r1 = r1 > (C_CLASSES - 1) ? (C_CLASSES - 1) : r1;
            v2f B;
            B.x = out[r0 * D_DIM + n0 + m];                  // B[k,n] = proto_sum row
            B.y = out[r1 * D_DIM + n0 + m];

            acc = __builtin_amdgcn_wmma_f32_16x16x4_f32(
                /*neg_a=*/false, A, /*neg_b=*/false, B,
                /*c_mod=*/(short)0, acc, /*reuse_a=*/false, /*reuse_b=*/false);
        }
#pragma unroll
        for (int r = 0; r < 8; ++r) {
            const int row = base + r + (hi ? 8 : 0);
            if (row < C_CLASSES)
                out[row * D_DIM + n0 + m] = acc[r];
        }
    }
}

// ---------------- Launcher ------------------------------------------------------------
extern "C" void kernel_launch(void* const* d_in, const int* in_sizes, int n_in,
                              void* d_out, int out_size, void* d_ws, size_t ws_size,
                              hipStream_t stream) {
    const float* emb    = (const float*)d_in[0];
    const int*   labels = (const int*)d_in[1];
    const float* conf   = (const float*)d_in[2];
    float* out     = (float*)d_out;
    float* confsum = (float*)d_ws;              // 1000 floats of scratch
    const int n = in_sizes[2];                  // N samples (confidences count)

    proto_zero_kernel<<<256, 256, 0, stream>>>(out, confsum);

    int blocks = (n + 7) / 8;                   // 8 waves/block, 1 sample/wave/iter
    if (blocks > 2048) blocks = 2048;
    if (blocks < 1) blocks = 1;
    proto_accum_kernel<<<blocks, 256, 0, stream>>>(emb, labels, conf, out, confsum, n);

    proto_norm_kernel<<<(C_CLASSES + 15) / 16, 256, 0, stream>>>(out, confsum);
}